// CheiralityLayer_76355928588465
// MI455X (gfx1250) — compile-verified
//
#include <hip/hip_runtime.h>
#include <math.h>

// ---------------------------------------------------------------------------
// CheiralityLayer: rho = (g . A V) * (n0+n1 - g . B W);  out = mean gelu(-rho)
// Pure streaming reduction: ~79 MB single-pass reads @ 23.3 TB/s => ~3.4us.
// Memory bound by ~20x over VALU; A/B are closed-form in (x,y), never
// materialized. Loads are non-temporal (TH=NT) so the one-pass stream does
// not rinse the 192MB L2. Wave reduction uses V_WMMA_F32_16X16X4_F32 with an
// all-ones B matrix (exact f32 RNE adds) + one ds_swizzle SWAPX16.
// ---------------------------------------------------------------------------

typedef __attribute__((ext_vector_type(2))) float v2f;
typedef __attribute__((ext_vector_type(4))) float v4f;
typedef __attribute__((ext_vector_type(8))) float v8f;

#define HWC      307200      // 480*640
#define IMG_W    640
#define NBATCH   16
#define NTOT     (NBATCH * HWC)   // 4,915,200 pixels
#define NBLOCKS  1024
#define NTHREADS 256
#define NWAVES   (NTHREADS / 32)

// Exact (non-approximate) GELU, f32.
__device__ __forceinline__ float gelu_exact(float z) {
    return 0.5f * z * (1.0f + erff(z * 0.70710678118654752440f));
}

// Single-instruction XOR-16 lane exchange: ds_swizzle group-of-32,
// offset = {xor_mask=0x10, or_mask=0x00, and_mask=0x1f} -> 0x401F (SWAPX16).
__device__ __forceinline__ float swap_x16(float v) {
    int iv = __builtin_bit_cast(int, v);
    iv = __builtin_amdgcn_ds_swizzle(iv, 0x401F);
    return __builtin_bit_cast(float, iv);
}

// Wave32 sum reduction via V_WMMA_F32_16X16X4_F32.
// A(16x4) layout: lanes 0-15 hold {K=0,K=1} in VGPR0/1, lanes 16-31 {K=2,K=3}.
// Put acc in component 0 (A[:,0] and A[:,2]), zeros elsewhere; B = all-ones:
//   D[m,n] = acc_m + acc_{m+16}  for every n.
// Lane l<16 holds D[0..7, l]; lane l>=16 holds D[8..15, l-16]. The per-lane
// component sum s is uniform within each half-wave; total = s + swap_x16(s).
__device__ __forceinline__ float wave_sum32(float acc) {
#if defined(__gfx1250__) && __has_builtin(__builtin_amdgcn_wmma_f32_16x16x4_f32)
    v2f a; a[0] = acc;  a[1] = 0.0f;
    v2f b; b[0] = 1.0f; b[1] = 1.0f;
    v8f c = {};
    c = __builtin_amdgcn_wmma_f32_16x16x4_f32(
        /*neg_a=*/false, a, /*neg_b=*/false, b,
        /*c_mod=*/(short)0, c, /*reuse_a=*/false, /*reuse_b=*/false);
    float s = ((c[0] + c[1]) + (c[2] + c[3])) + ((c[4] + c[5]) + (c[6] + c[7]));
    s += swap_x16(s);
    return s;
#else
    for (int off = 16; off > 0; off >>= 1) acc += __shfl_xor(acc, off, 32);
    return acc;
#endif
}

__global__ __launch_bounds__(NTHREADS)
void cheirality_main(const float* __restrict__ pose,
                     const float* __restrict__ grad,
                     const float* __restrict__ nflow,
                     float* __restrict__ partials) {
    __shared__ float sp[NBATCH * 6];
    __shared__ float wpart[NWAVES];

    if (threadIdx.x < NBATCH * 6) sp[threadIdx.x] = pose[threadIdx.x];
    __syncthreads();

    const int n4     = NTOT / 4;                 // float4 granules
    const int stride = NBLOCKS * NTHREADS;

    float acc = 0.0f;
    for (int i = blockIdx.x * NTHREADS + threadIdx.x; i < n4; i += stride) {
        const int pix = i * 4;                   // flat pixel index in [0, NTOT)
        const int b   = pix / HWC;               // HWC % 4 == 0 -> all 4 pixels same b
        const int p   = pix - b * HWC;
        const int y   = p / IMG_W;               // IMG_W % 4 == 0 -> same row
        const int x   = p - y * IMG_W;
        const int gb  = b * (2 * HWC) + p;

        // Hint the next grid-stride line toward GL2 (speculative; dropped if OOB).
        if (i + stride < n4) {
            __builtin_prefetch(grad + gb + 4 * stride, 0, 1);
            __builtin_prefetch(nflow + gb + 4 * stride, 0, 1);
        }

        // Single-use streaming data: non-temporal loads (global_load_b128, TH=NT).
        v4f g0v = __builtin_nontemporal_load((const v4f*)(grad  + gb));
        v4f g1v = __builtin_nontemporal_load((const v4f*)(grad  + gb + HWC));
        v4f n0v = __builtin_nontemporal_load((const v4f*)(nflow + gb));
        v4f n1v = __builtin_nontemporal_load((const v4f*)(nflow + gb + HWC));

        const float* pb = sp + b * 6;
        const float V0 = pb[0], V1 = pb[1], V2 = pb[2];
        const float W0 = pb[3], W1 = pb[4], W2 = pb[5];

        const float yf   = (float)y;
        const float AV1  = yf * V2 - V1;                 // row-constant
        const float yW2  = yf * W2;
        const float BW1c = (yf * yf + 1.0f) * W0;        // row-constant part

#pragma unroll
        for (int j = 0; j < 4; ++j) {
            const float xf  = (float)(x + j);
            const float xy  = xf * yf;
            const float AV0 = xf * V2 - V0;
            const float BW0 = xy * W0 - (xf * xf + 1.0f) * W1 + yW2;
            const float BW1 = BW1c - xy * W1 - xf * W2;
            const float dA  = g0v[j] * AV0 + g1v[j] * AV1;
            const float dB  = g0v[j] * BW0 + g1v[j] * BW1;
            const float rho = dA * ((n0v[j] + n1v[j]) - dB);
            acc += gelu_exact(-rho);
        }
    }

    // Uniform control flow from here: EXEC is all ones for the WMMA.
    const float ws = wave_sum32(acc);
    const int lane = threadIdx.x & 31;
    const int wid  = threadIdx.x >> 5;
    if (lane == 0) wpart[wid] = ws;
    __syncthreads();
    if (threadIdx.x == 0) {
        float t = 0.0f;
#pragma unroll
        for (int w = 0; w < NWAVES; ++w) t += wpart[w];
        partials[blockIdx.x] = t;
    }
}

__global__ __launch_bounds__(NTHREADS)
void cheirality_finish(const float* __restrict__ partials,
                       float* __restrict__ out) {
    __shared__ float wpart[NWAVES];
    float acc = 0.0f;
    for (int i = threadIdx.x; i < NBLOCKS; i += NTHREADS) acc += partials[i];
    const float ws = wave_sum32(acc);
    const int lane = threadIdx.x & 31;
    const int wid  = threadIdx.x >> 5;
    if (lane == 0) wpart[wid] = ws;
    __syncthreads();
    if (threadIdx.x == 0) {
        float s = 0.0f;
#pragma unroll
        for (int w = 0; w < NWAVES; ++w) s += wpart[w];
        out[0] = s * (1.0f / (float)NTOT);
    }
}

extern "C" void kernel_launch(void* const* d_in, const int* in_sizes, int n_in,
                              void* d_out, int out_size, void* d_ws, size_t ws_size,
                              hipStream_t stream) {
    const float* pose  = (const float*)d_in[0];   // (16, 6)
    const float* grad  = (const float*)d_in[1];   // (16, 2, 480, 640)
    const float* nflow = (const float*)d_in[2];   // (16, 2, 480, 640)
    float* partials    = (float*)d_ws;            // NBLOCKS floats, rewritten every call

    cheirality_main<<<NBLOCKS, NTHREADS, 0, stream>>>(pose, grad, nflow, partials);
    cheirality_finish<<<1, NTHREADS, 0, stream>>>(partials, (float*)d_out);

    (void)in_sizes; (void)n_in; (void)out_size; (void)ws_size;
}